// HungarianMatcher_15324443312205
// MI455X (gfx1250) — compile-verified
//
#include <hip/hip_runtime.h>
#include <math.h>

typedef __attribute__((ext_vector_type(2))) float v2f;
typedef __attribute__((ext_vector_type(8))) float v8f;

#define BATCH 32
#define N_OUT 512     // pred count  (LAP columns m)
#define M_TGT 200     // gt count    (LAP rows n)
#define DDIM  768
#define MTILES ((M_TGT + 15) / 16)   // 13 (rows 200..207 padded, stores guarded)
#define NTILES (N_OUT / 16)          // 32
#define BIGF  1e9f

// ---------------------------------------------------------------------------
// Kernel 1: squared row norms. One wave per row (targets then outputs).
// ---------------------------------------------------------------------------
__global__ void __launch_bounds__(256)
norms_kernel(const float* __restrict__ outputs, const float* __restrict__ targets,
             float* __restrict__ o2, float* __restrict__ t2) {
  int wave = (blockIdx.x * blockDim.x + threadIdx.x) >> 5;
  int lane = threadIdx.x & 31;
  const int tgtRows = BATCH * M_TGT;   // 6400
  const int outRows = BATCH * N_OUT;   // 16384
  if (wave >= tgtRows + outRows) return;
  const float* row;
  float* dst;
  if (wave < tgtRows) { row = targets + (size_t)wave * DDIM; dst = t2 + wave; }
  else { int w = wave - tgtRows; row = outputs + (size_t)w * DDIM; dst = o2 + w; }
  float s = 0.f;
  for (int k = lane; k < DDIM; k += 32) { float x = row[k]; s += x * x; }
  for (int off = 16; off > 0; off >>= 1) s += __shfl_xor(s, off, 32);
  if (lane == 0) *dst = s;
}

// ---------------------------------------------------------------------------
// Kernel 2: cost^T[b][m_gt][n_pred] = ||targets[b,m]-outputs[b,n]|| via f32 WMMA.
// One wave computes one 16x16 tile, K-loop over 768 in steps of 4 (2 accums).
// A-frag layout (16x4 f32): lane<16 -> M=lane, K=k+{0,1}; lane>=16 -> K=k+{2,3}.
// B-frag layout (4x16 f32): lane<16 -> N=lane, K=k+{0,1}; lane>=16 -> K=k+{2,3}.
// Both are contiguous float2 loads from row-major targets/outputs.
// ---------------------------------------------------------------------------
__global__ void __launch_bounds__(256)
cost_wmma_kernel(const float* __restrict__ outputs, const float* __restrict__ targets,
                 const float* __restrict__ o2, const float* __restrict__ t2,
                 float* __restrict__ cost) {
  int wave = (blockIdx.x * blockDim.x + threadIdx.x) >> 5;
  int lane = threadIdx.x & 31;
  const int tilesPerBatch = MTILES * NTILES;  // 416
  int b   = wave / tilesPerBatch;
  int rem = wave - b * tilesPerBatch;
  int mt = rem / NTILES;
  int nt = rem - mt * NTILES;

  int hi = lane >> 4;        // 0 or 1 -> K sub-offset
  int lq = lane & 15;

  int mrow = mt * 16 + lq; if (mrow > M_TGT - 1) mrow = M_TGT - 1;  // clamp pad rows
  int ncol = nt * 16 + lq;

  const float* aptr = targets + ((size_t)b * M_TGT + mrow) * DDIM + 2 * hi;
  const float* bptr = outputs + ((size_t)b * N_OUT + ncol) * DDIM + 2 * hi;

  v8f acc0 = {};
  v8f acc1 = {};
#pragma unroll 4
  for (int k = 0; k < DDIM; k += 8) {
    v2f a0 = *(const v2f*)(aptr + k);
    v2f b0 = *(const v2f*)(bptr + k);
    v2f a1 = *(const v2f*)(aptr + k + 4);
    v2f b1 = *(const v2f*)(bptr + k + 4);
    acc0 = __builtin_amdgcn_wmma_f32_16x16x4_f32(false, a0, false, b0,
                                                 (short)0, acc0, false, false);
    acc1 = __builtin_amdgcn_wmma_f32_16x16x4_f32(false, a1, false, b1,
                                                 (short)0, acc1, false, false);
  }

  float o2v = o2[(size_t)b * N_OUT + ncol];
  // C/D layout: lane, vgpr r -> (M = r + 8*(lane>>4), N = lane&15)
#pragma unroll
  for (int r = 0; r < 8; ++r) {
    int row = mt * 16 + r + 8 * hi;
    if (row < M_TGT) {
      float t2v = t2[(size_t)b * M_TGT + row];
      float d2 = t2v + o2v - 2.0f * (acc0[r] + acc1[r]);
      cost[((size_t)b * M_TGT + row) * N_OUT + nt * 16 + lq] =
          sqrtf(fmaxf(d2, 0.0f));
    }
  }
}

// ---------------------------------------------------------------------------
// Kernel 3: Jonker-Volgenant LAP, one wave32 per batch. n=200 rows, m=512 cols.
// State in LDS; cost rows stream from L2 (13MB fits easily in 192MB L2).
// ---------------------------------------------------------------------------
__global__ void __launch_bounds__(32)
lap_kernel(const float* __restrict__ cost, int* __restrict__ out) {
  __shared__ float v[N_OUT + 1];
  __shared__ float minv[N_OUT + 1];
  __shared__ float u[M_TGT + 1];
  __shared__ int   way[N_OUT + 1];
  __shared__ int   p[N_OUT + 1];
  __shared__ int   used[N_OUT + 1];
  __shared__ int   g2p[M_TGT];

  const int b = blockIdx.x;
  const int lane = threadIdx.x;
  const float* cb = cost + (size_t)b * M_TGT * N_OUT;

  for (int j = lane; j <= N_OUT; j += 32) { v[j] = 0.f; p[j] = 0; }
  for (int i = lane; i <= M_TGT; i += 32) u[i] = 0.f;
  __syncthreads();

  for (int i = 1; i <= M_TGT; ++i) {
    if (lane == 0) p[0] = i;
    for (int j = lane; j <= N_OUT; j += 32) { minv[j] = BIGF; used[j] = 0; way[j] = 0; }
    __syncthreads();

    int j0 = 0;
    for (int guard = 0; guard < 4096; ++guard) {
      if (lane == 0) used[j0] = 1;
      __syncthreads();
      int   i0  = p[j0];
      float ui0 = u[i0];
      const float* arow = cb + (size_t)(i0 - 1) * N_OUT;

      // relax reduced costs over unused columns (16 per lane, coalesced)
#pragma unroll
      for (int it = 0; it < 16; ++it) {
        int j = 1 + lane + (it << 5);
        if (!used[j]) {
          float cur = arow[j - 1] - ui0 - v[j];
          if (cur < minv[j]) { minv[j] = cur; way[j] = j0; }
        }
      }
      __syncthreads();

      // argmin over unused columns, tie -> lowest index (matches jnp.argmin)
      float bv = BIGF; int bj = 1 << 30;
#pragma unroll
      for (int it = 0; it < 16; ++it) {
        int j = 1 + lane + (it << 5);
        float mv = used[j] ? BIGF : minv[j];
        if (mv < bv) { bv = mv; bj = j; }
      }
      for (int off = 16; off > 0; off >>= 1) {
        float ov = __shfl_xor(bv, off, 32);
        int   oj = __shfl_xor(bj, off, 32);
        if (ov < bv || (ov == bv && oj < bj)) { bv = ov; bj = oj; }
      }
      float delta = bv;
      int   j1    = bj;

      // dual update; u[p[j]] targets are distinct across used j -> race-free
#pragma unroll
      for (int it = 0; it < 16; ++it) {
        int j = 1 + lane + (it << 5);
        if (used[j]) { u[p[j]] += delta; v[j] -= delta; }
        else         { minv[j] -= delta; }
      }
      if (lane == 0) u[p[0]] += delta;  // virtual column 0 (current row i)
      __syncthreads();

      j0 = j1;
      if (p[j0] == 0) break;
    }

    // backtrack augmenting path
    if (lane == 0) {
      int jj = j0;
      while (jj != 0) { int jp = way[jj]; p[jj] = p[jp]; jj = jp; }
    }
    __syncthreads();
  }

  // invert matching: gt row r -> pred col
  for (int j = 1 + lane; j <= N_OUT; j += 32) {
    int r = p[j];
    if (r > 0) g2p[r - 1] = j - 1;
  }
  __syncthreads();

  // scipy-style sort by pred index: rank = #smaller (cols are distinct)
  int* ob = out + b * 2 * M_TGT;
  for (int r = lane; r < M_TGT; r += 32) {
    int val = g2p[r];
    int cnt = 0;
    for (int r2 = 0; r2 < M_TGT; ++r2) cnt += (g2p[r2] < val) ? 1 : 0;
    ob[cnt]         = val;  // index_i (pred), ascending
    ob[M_TGT + cnt] = r;    // index_j (gt)
  }
}

// ---------------------------------------------------------------------------
extern "C" void kernel_launch(void* const* d_in, const int* in_sizes, int n_in,
                              void* d_out, int out_size, void* d_ws, size_t ws_size,
                              hipStream_t stream) {
  (void)in_sizes; (void)n_in; (void)out_size; (void)ws_size;
  const float* outputs = (const float*)d_in[0];  // [32,512,768] f32
  const float* targets = (const float*)d_in[1];  // [32,200,768] f32

  float* cost = (float*)d_ws;                               // [32][200][512]
  float* o2   = cost + (size_t)BATCH * M_TGT * N_OUT;       // [32][512]
  float* t2   = o2 + (size_t)BATCH * N_OUT;                 // [32][200]
  int*   out  = (int*)d_out;                                // [32][2][200]

  {
    int totalWaves = BATCH * (M_TGT + N_OUT);               // 22784
    int blocks = (totalWaves * 32 + 255) / 256;             // 2848
    norms_kernel<<<blocks, 256, 0, stream>>>(outputs, targets, o2, t2);
  }
  {
    int totalWaves = BATCH * MTILES * NTILES;               // 13312
    cost_wmma_kernel<<<totalWaves / 8, 256, 0, stream>>>(outputs, targets,
                                                         o2, t2, cost);
  }
  lap_kernel<<<BATCH, 32, 0, stream>>>(cost, out);
}